// GMMNet_U_12773232739128
// MI455X (gfx1250) — compile-verified
//
#include <hip/hip_runtime.h>
#include <math.h>

typedef __attribute__((ext_vector_type(2))) float v2f;
typedef __attribute__((ext_vector_type(8))) float v8f;

#define N_NODES 50000
#define N_EDGES 400000
#define ROW_TILES (N_NODES / 16)   // 3125 exactly

// ---------------------------------------------------------------- utilities
__global__ void fill_kernel(float* __restrict__ p, size_t n, float v) {
    size_t i = (size_t)blockIdx.x * blockDim.x + threadIdx.x;
    if (i < n) p[i] = v;
}

__global__ void degree_kernel(const int* __restrict__ dst, float* __restrict__ cnt) {
    int e = blockIdx.x * blockDim.x + threadIdx.x;
    if (e < N_EDGES) atomicAdd(&cnt[dst[e]], 1.0f);
}

// Pack fused weights [g | root | 0] into K-pair-interleaved layout:
//   Bpack(float2)[kpair * colsPad + c] = { W[2*kpair][c], W[2*kpair+1][c] }
// so a WMMA B-fragment is a single aligned b64 load.
__global__ void pack_weights_kernel(const float* __restrict__ g,
                                    const float* __restrict__ root,
                                    float2* __restrict__ Bp,
                                    int cin, int cout, int colsPad) {
    int i = blockIdx.x * blockDim.x + threadIdx.x;
    int nPairs = cin >> 1;
    int n = nPairs * colsPad;
    if (i >= n) return;
    int kp = i / colsPad;
    int c  = i - kp * colsPad;
    float2 v = make_float2(0.0f, 0.0f);
    if (c < 3 * cout) {
        v.x = g[(2 * kp + 0) * 3 * cout + c];
        v.y = g[(2 * kp + 1) * 3 * cout + c];
    } else if (c < 4 * cout) {
        int cc = c - 3 * cout;
        v.x = root[(2 * kp + 0) * cout + cc];
        v.y = root[(2 * kp + 1) * cout + cc];
    }
    Bp[i] = v;
}

// ---------------------------------------------------------------- WMMA GEMM
// Y(N x colsPad) = A(N x cin) * W(cin x colsPad) via V_WMMA_F32_16X16X4_F32.
// Each block stages its 64-column weight panel (all cin rows, pair-packed) in
// LDS once; 8 waves then each compute a 16x64 output strip, feeding WMMA from
// ds_load_b64 B-fragments and streaming b64 A-fragments from global.
__global__ __launch_bounds__(256) void gemm_wmma_kernel(
    const float* __restrict__ A, const float2* __restrict__ Bp,
    float* __restrict__ Y, int cin, int colsPad) {
    extern __shared__ char smemRaw[];
    float2* sB = (float2*)smemRaw;               // nPairs x 64 float2

    const int colBase = blockIdx.y * 64;
    const int nPairs  = cin >> 1;

    // Cooperative LDS fill: panel = all K pairs for 64 columns.
    for (int i = threadIdx.x; i < nPairs * 64; i += 256) {
        int kp = i >> 6;
        int c  = i & 63;
        sB[i] = Bp[(size_t)kp * colsPad + colBase + c];
    }
    __syncthreads();

    const int lane = threadIdx.x & 31;
    const int wave = threadIdx.x >> 5;
    const int rowTile = blockIdx.x * 8 + wave;
    if (rowTile < ROW_TILES) {                   // wave-uniform; EXEC all-1s inside
        const int m    = lane & 15;              // row in tile (A) / col in tile (B,C)
        const int kSel = lane >> 4;              // K pair {0,1} vs {2,3}

        v8f c0 = {}, c1 = {}, c2 = {}, c3 = {};
        const float2* __restrict__ arow =
            (const float2*)(A + (size_t)(rowTile * 16 + m) * cin);

        for (int k0 = 0; k0 < cin; k0 += 4) {
            float2 at = arow[(k0 >> 1) + kSel];
            v2f a; a.x = at.x; a.y = at.y;
            const float2* __restrict__ bpnl = sB + ((k0 >> 1) + kSel) * 64 + m;
            float2 t0 = bpnl[0], t1 = bpnl[16], t2 = bpnl[32], t3 = bpnl[48];
            v2f b0; b0.x = t0.x; b0.y = t0.y;
            v2f b1; b1.x = t1.x; b1.y = t1.y;
            v2f b2; b2.x = t2.x; b2.y = t2.y;
            v2f b3; b3.x = t3.x; b3.y = t3.y;
            c0 = __builtin_amdgcn_wmma_f32_16x16x4_f32(false, a, false, b0, (short)0, c0, false, false);
            c1 = __builtin_amdgcn_wmma_f32_16x16x4_f32(false, a, false, b1, (short)0, c1, false, false);
            c2 = __builtin_amdgcn_wmma_f32_16x16x4_f32(false, a, false, b2, (short)0, c2, false, false);
            c3 = __builtin_amdgcn_wmma_f32_16x16x4_f32(false, a, false, b3, (short)0, c3, false, false);
        }

        // C/D layout: VGPR v -> row v (lanes 0-15) or v+8 (lanes 16-31), col = lane&15
        const int rowOff = rowTile * 16 + kSel * 8;
#pragma unroll
        for (int v = 0; v < 8; ++v) {
            float* __restrict__ yr = Y + (size_t)(rowOff + v) * colsPad + colBase + m;
            yr[0]  = c0[v];
            yr[16] = c1[v];
            yr[32] = c2[v];
            yr[48] = c3[v];
        }
    }
}

// ---------------------------------------------------------------- edge pass
// One wave per edge: gaussian mixture weights from edge_attr, coalesced gather
// of xg[src] row (b128 when cout is a multiple of 128), atomic scatter to dst.
__global__ __launch_bounds__(256) void edge_msg_kernel(
    const float* __restrict__ xg, const int* __restrict__ src,
    const int* __restrict__ dst, const float* __restrict__ ea,
    const float* __restrict__ mu, const float* __restrict__ sigma,
    float* __restrict__ agg, int cout, int colsPad) {
    const int e    = (int)(((size_t)blockIdx.x * blockDim.x + threadIdx.x) >> 5);
    const int lane = threadIdx.x & 31;
    if (e >= N_EDGES) return;                    // wave-uniform
    const int s = src[e];
    const int d = dst[e];
    const float u = ea[e];
    float w0, w1, w2;
    {
        float d0 = u - mu[0], d1 = u - mu[1], d2 = u - mu[2];
        float s0 = sigma[0],  s1 = sigma[1],  s2 = sigma[2];
        w0 = expf(-0.5f * d0 * d0 / (1e-15f + s0 * s0));
        w1 = expf(-0.5f * d1 * d1 / (1e-15f + s1 * s1));
        w2 = expf(-0.5f * d2 * d2 / (1e-15f + s2 * s2));
    }
    const float* __restrict__ row = xg + (size_t)s * colsPad;
    float* __restrict__ arow = agg + (size_t)d * cout;

    if ((cout & 127) == 0) {                     // cout = 128 or 256: b128 path
        const float4* __restrict__ r0 = (const float4*)row;
        const float4* __restrict__ r1 = (const float4*)(row + cout);
        const float4* __restrict__ r2 = (const float4*)(row + 2 * cout);
        int n4 = cout >> 2;
        for (int q = lane; q < n4; q += 32) {
            float4 a0 = r0[q], a1 = r1[q], a2 = r2[q];
            float4 msg;
            msg.x = w0 * a0.x + w1 * a1.x + w2 * a2.x;
            msg.y = w0 * a0.y + w1 * a1.y + w2 * a2.y;
            msg.z = w0 * a0.z + w1 * a1.z + w2 * a2.z;
            msg.w = w0 * a0.w + w1 * a1.w + w2 * a2.w;
            float* ap = arow + q * 4;
            atomicAdd(&ap[0], msg.x);
            atomicAdd(&ap[1], msg.y);
            atomicAdd(&ap[2], msg.z);
            atomicAdd(&ap[3], msg.w);
        }
    } else {
        for (int f = lane; f < cout; f += 32) {
            float msg = w0 * row[f] + w1 * row[cout + f] + w2 * row[2 * cout + f];
            atomicAdd(&arow[f], msg);
        }
    }
}

// ---------------------------------------------------------------- finalize
// out = agg/max(cnt,1) + root_part + bias  (+skip)  (gelu?)  -> xout/skipout
__global__ void finalize_kernel(
    const float* __restrict__ agg, const float* __restrict__ y,
    const float* __restrict__ bias, const float* __restrict__ cnt,
    const float* __restrict__ skip, float* __restrict__ xout,
    float* __restrict__ skipout, int cout, int colsPad, int doAct) {
    size_t i = (size_t)blockIdx.x * blockDim.x + threadIdx.x;
    size_t n_tot = (size_t)N_NODES * cout;
    if (i >= n_tot) return;
    int n = (int)(i / cout);
    int f = (int)(i - (size_t)n * cout);
    float v = agg[i] / fmaxf(cnt[n], 1.0f)
            + y[(size_t)n * colsPad + 3 * cout + f]
            + bias[f];
    if (skip) v += skip[i];
    if (doAct) v = 0.5f * v * (1.0f + erff(v * 0.7071067811865476f));
    xout[i] = v;
    if (skipout) skipout[i] = v;
}

// ---------------------------------------------------------------- launcher
extern "C" void kernel_launch(void* const* d_in, const int* in_sizes, int n_in,
                              void* d_out, int out_size, void* d_ws, size_t ws_size,
                              hipStream_t stream) {
    (void)in_sizes; (void)n_in; (void)out_size; (void)ws_size;

    const float* x_in = (const float*)d_in[0];
    const int*   ei   = (const int*)d_in[1];      // (2, E)
    const float* ea   = (const float*)d_in[2];    // (E, 1)
    const int* src = ei;
    const int* dst = ei + N_EDGES;

    static const int cins[8]  = {4, 32, 64, 128, 256, 128, 64, 32};
    static const int couts[8] = {32, 64, 128, 256, 128, 64, 32, 1};

    // Workspace carve-up (floats)
    float* W      = (float*)d_ws;
    float* cnt    = W;                                  // N
    float* xbuf   = cnt   + N_NODES;                    // N*256
    float* ybuf   = xbuf  + (size_t)N_NODES * 256;      // N*1024
    float* aggbuf = ybuf  + (size_t)N_NODES * 1024;     // N*256
    float* h0     = aggbuf + (size_t)N_NODES * 256;     // N*32
    float* h1     = h0    + (size_t)N_NODES * 32;       // N*64
    float* h2     = h1    + (size_t)N_NODES * 64;       // N*128
    float2* Bp    = (float2*)(h2 + (size_t)N_NODES * 128);  // (256/2)*1024 float2

    // Degree (layer-invariant)
    {
        int blocks = (N_NODES + 255) / 256;
        fill_kernel<<<blocks, 256, 0, stream>>>(cnt, (size_t)N_NODES, 0.0f);
        degree_kernel<<<(N_EDGES + 255) / 256, 256, 0, stream>>>(dst, cnt);
    }

    const float* curx = x_in;
    for (int L = 0; L < 8; ++L) {
        const int cin  = cins[L];
        const int cout = couts[L];
        const int colsPad = ((4 * cout + 63) / 64) * 64;

        const float* g     = (const float*)d_in[3 + 5 * L + 0];
        const float* mu    = (const float*)d_in[3 + 5 * L + 1];
        const float* sigma = (const float*)d_in[3 + 5 * L + 2];
        const float* root  = (const float*)d_in[3 + 5 * L + 3];
        const float* bias  = (const float*)d_in[3 + 5 * L + 4];

        // 1) pack fused [g | root | 0] weights, K-pair interleaved
        int npack = (cin >> 1) * colsPad;
        pack_weights_kernel<<<(npack + 255) / 256, 256, 0, stream>>>(
            g, root, Bp, cin, cout, colsPad);

        // 2) WMMA GEMM: ybuf = curx @ W   (LDS-staged weight panel per block)
        dim3 gg((ROW_TILES + 7) / 8, colsPad / 64);
        size_t smem = (size_t)(cin >> 1) * 64 * sizeof(float2);  // <= 64 KB
        gemm_wmma_kernel<<<gg, 256, smem, stream>>>(curx, Bp, ybuf, cin, colsPad);

        // 3) zero agg, then edge gather/scatter
        size_t na = (size_t)N_NODES * cout;
        fill_kernel<<<(int)((na + 255) / 256), 256, 0, stream>>>(aggbuf, na, 0.0f);
        edge_msg_kernel<<<(N_EDGES + 7) / 8, 256, 0, stream>>>(
            ybuf, src, dst, ea, mu, sigma, aggbuf, cout, colsPad);

        // 4) finalize: mean + root + bias (+skip) (+gelu)
        const float* skip = (L == 4) ? h2 : (L == 5) ? h1 : (L == 6) ? h0 : nullptr;
        float* skipout    = (L == 0) ? h0 : (L == 1) ? h1 : (L == 2) ? h2 : nullptr;
        float* xout       = (L == 7) ? (float*)d_out : xbuf;
        int doAct         = (L < 7) ? 1 : 0;
        finalize_kernel<<<(int)((na + 255) / 256), 256, 0, stream>>>(
            aggbuf, ybuf, bias, cnt, skip, xout, skipout, cout, colsPad, doAct);

        curx = xbuf;
    }
}